// Ngrammer_78778290143891
// MI455X (gfx1250) — compile-verified
//
#include <hip/hip_runtime.h>
#include <math.h>

typedef __attribute__((ext_vector_type(2))) float vf2;
typedef __attribute__((ext_vector_type(8))) float v8f;

#define BATCH 8
#define SEQ   4096
#define NHEAD 16
#define HDIM  64
#define NGDIM 8
#define UNIV  32000
#define NGV   196608
#define LNEPS 1e-5f

#define TILE_T 16
#define TILE_H 8
#define ROWF   (TILE_H * HDIM + 4)   /* 516 floats: stride%64==4 -> conflict-free LDS */

struct HashConsts {
    unsigned int       p[NHEAD];     // primes
    unsigned long long m[NHEAD];     // magic: floor(2^64/p)+1
};

__global__ __launch_bounds__(256) void ngrammer_kernel(
    const float* __restrict__ embeds,        // (B, N, 1024) f32
    const long long* __restrict__ cids,      // (B, N, 16)   i64
    const float* __restrict__ tables,        // (16, NGV, 8) f32
    const float* __restrict__ g_ng,          // (16, 8)
    const float* __restrict__ b_ng,          // (16, 8)
    const float* __restrict__ g_em,          // (16, 64)
    const float* __restrict__ b_em,          // (16, 64)
    float* __restrict__ out,                 // (B, N, 1024) f32
    HashConsts C)
{
    __shared__ float  sX[TILE_T * ROWF];         // ~33 KB staged tile
    __shared__ float2 sStat[TILE_H][TILE_T];     // (mean, 1/(sigma+eps)) per (head, token)

    const int tid  = threadIdx.x;
    const int lane = tid & 31;
    const int wave = tid >> 5;

    const int bid  = blockIdx.x;                 // 0..4095
    const int hh   = bid & 1;
    const int tile = (bid >> 1) & (SEQ / TILE_T - 1);
    const int b    = bid >> 9;
    const int h0   = hh * TILE_H;
    const int n0   = tile * TILE_T;

    const int hl = wave;                         // this wave's head (local)
    const int h  = h0 + hl;                      // this wave's head (global)

    // ---------------- Phase 1: coalesced global -> LDS stage ----------------
    #pragma unroll
    for (int it = 0; it < 8; ++it) {
        int idx = it * 256 + tid;                // 0..2047 float4 slots
        int row = idx >> 7;                      // 128 float4 per row-half
        int c4  = idx & 127;
        const float4 v = *(const float4*)(embeds
            + (size_t)(b * SEQ + n0 + row) * (NHEAD * HDIM)
            + h0 * HDIM + c4 * 4);
        *(float4*)(&sX[row * ROWF + c4 * 4]) = v;
    }
    __syncthreads();

    // ----- Phase 2: f32 WMMA layernorm statistics (sum, sumsq) per head -----
    {
        const int M    = lane & 15;              // A-matrix row = token
        const int half = lane >> 4;              // K pair select

        // A fragment (16x4 f32): lane M holds K = {2*half, 2*half+1}
        const float* base = &sX[M * ROWF + hl * HDIM + half * 2];

        vf2 ones2; ones2[0] = 1.0f; ones2[1] = 1.0f;
        v8f accS = {};
        v8f accQ = {};

        #pragma unroll
        for (int c = 0; c < 16; ++c) {           // 16 K=4 chunks cover D=64
            const vf2 a = *(const vf2*)(base + 4 * c);
            accS = __builtin_amdgcn_wmma_f32_16x16x4_f32(false, a, false, ones2,
                                                         (short)0, accS, false, false);
            const vf2 q = a * a;                 // v_pk_mul_f32
            accQ = __builtin_amdgcn_wmma_f32_16x16x4_f32(false, q, false, ones2,
                                                         (short)0, accQ, false, false);
        }

        // D layout: VGPR r, lane 0 -> token r; lane 16 -> token 8+r
        if ((lane & 15) == 0) {
            const int tb = half * 8;
            #pragma unroll
            for (int r = 0; r < 8; ++r) {
                float mean = accS[r] * (1.0f / HDIM);
                float var  = fmaxf(accQ[r] * (1.0f / HDIM) - mean * mean, 0.0f);
                float2 st; st.x = mean; st.y = 1.0f / (sqrtf(var) + LNEPS);
                sStat[hl][tb + r] = st;
            }
        }
        // No block barrier needed: stats are produced and consumed by the same
        // wave; LDS ops within a wave are in-order (DScnt waits inserted).
    }

    // ---------- Phase 3: normalize + ngram hash/gather/LN + store ----------
    // This wave handles its own head `h` for all 16 tokens: every per-head
    // constant is loop-invariant, addresses are simple inductions.
    const int ch = lane * 2;
    const bool doEmb = (ch < HDIM - NGDIM);
    const bool doNg  = (lane < NGDIM);

    float2 geh, beh;
    if (doEmb) {
        geh = *(const float2*)(g_em + h * HDIM + ch);
        beh = *(const float2*)(b_em + h * HDIM + ch);
    }
    float gnh = 0.f, bnh = 0.f;
    long long prev = 0;
    if (doNg) {
        gnh = g_ng[h * NGDIM + lane];
        bnh = b_ng[h * NGDIM + lane];
        if (n0 != 0)
            prev = cids[(size_t)(b * SEQ + n0 - 1) * NHEAD + h];
    }
    const unsigned long long aa  = (unsigned long long)(h + 1);
    const unsigned long long mag = C.m[h];
    const unsigned long long pr  = (unsigned long long)C.p[h];
    const float* tabh = tables + (size_t)h * NGV * NGDIM;

    const float*     xp = &sX[hl * HDIM];
    float*           op = out  + (size_t)(b * SEQ + n0) * (NHEAD * HDIM) + h * HDIM;
    const long long* cp = cids + (size_t)(b * SEQ + n0) * NHEAD + h;

    #pragma unroll 4
    for (int t = 0; t < 16; ++t) {
        const float2 st = sStat[hl][t];          // (mean, inv)

        if (doEmb) {                             // channels 0..55: normed embeds
            const float2 x = *(const float2*)(xp + ch);
            float2 o;
            o.x = (x.x - st.x) * st.y * geh.x + beh.x;
            o.y = (x.y - st.x) * st.y * geh.y + beh.y;
            *(float2*)(op + ch) = o;
        }

        if (doNg) {                              // channels 56..63: normed ngram
            const long long id = *cp;
            const unsigned long long v =
                (unsigned long long)(id + prev * (long long)UNIV) * aa + aa;
            prev = id;                           // carry bigram state

            // exact v % p via magic multiply (v < 2^34)
            const unsigned long long q64 = __umul64hi(v, mag);
            unsigned int r = (unsigned int)(v - q64 * pr);
            if (r >= (unsigned int)NGV) r -= (unsigned int)NGV;   // prime < 2*NGV

            const float tv = tabh[(size_t)r * NGDIM + lane];
            float s = tv, q = tv * tv;
            #pragma unroll
            for (int m = 1; m < NGDIM; m <<= 1) {  // 8-wide reduce within wave
                s += __shfl_xor(s, m, 32);
                q += __shfl_xor(q, m, 32);
            }
            const float mn = s * (1.0f / NGDIM);
            const float vr = fmaxf(q * (1.0f / NGDIM) - mn * mn, 0.0f);
            const float iv = 1.0f / (sqrtf(vr) + LNEPS);
            op[(HDIM - NGDIM) + lane] = (tv - mn) * iv * gnh + bnh;
        }

        xp += ROWF;
        op += NHEAD * HDIM;
        cp += NHEAD;
    }
}

// Host: first 16 primes above NGV (matches sympy.primerange(NGV+1, 2*NGV) head)
// plus magic reciprocals for exact device-side modulo.
static HashConsts compute_consts() {
    HashConsts C;
    int cnt = 0;
    for (long long c = (long long)NGV + 1; cnt < NHEAD; ++c) {
        if ((c & 1) == 0) continue;
        bool prime = true;
        for (long long d = 3; d * d <= c; d += 2)
            if (c % d == 0) { prime = false; break; }
        if (prime) {
            C.p[cnt] = (unsigned int)c;
            C.m[cnt] = (~0ull) / (unsigned long long)c + 1ull; // floor(2^64/p)+1
            ++cnt;
        }
    }
    return C;
}

extern "C" void kernel_launch(void* const* d_in, const int* in_sizes, int n_in,
                              void* d_out, int out_size, void* d_ws, size_t ws_size,
                              hipStream_t stream) {
    (void)in_sizes; (void)n_in; (void)out_size; (void)d_ws; (void)ws_size;
    const float*     embeds = (const float*)    d_in[0];
    const long long* cids   = (const long long*)d_in[1];   // jnp.int64
    const float*     tables = (const float*)    d_in[2];
    const float*     g_ng   = (const float*)    d_in[3];
    const float*     b_ng   = (const float*)    d_in[4];
    const float*     g_em   = (const float*)    d_in[5];
    const float*     b_em   = (const float*)    d_in[6];

    const HashConsts C = compute_consts();     // deterministic, cheap, capture-safe

    const dim3 grid(BATCH * (SEQ / TILE_T) * (NHEAD / TILE_H));  // 4096 blocks
    const dim3 block(256);
    ngrammer_kernel<<<grid, block, 0, stream>>>(embeds, cids, tables,
                                                g_ng, b_ng, g_em, b_em,
                                                (float*)d_out, C);
}